// CustomTorchModel_27212912787871
// MI455X (gfx1250) — compile-verified
//
#include <hip/hip_runtime.h>

typedef __attribute__((ext_vector_type(2))) float v2f;
typedef __attribute__((ext_vector_type(8))) float v8f;

#define B_TOT   4096
#define NGEN    512
#define FGEN    64
#define EMB     16
#define OUT_STRIDE (2*NGEN + 1)    // 1025
#define WAVES   8                  // waves per block (wave32)
#define TILE_M  16                 // batch rows per block

__global__ __launch_bounds__(256) void fused_policy_value_kernel(
    const float* __restrict__ x,      // [B, N, F]
    const float* __restrict__ Wg,     // [F, EMB]
    const float* __restrict__ bg,     // [EMB]
    const float* __restrict__ Wv,     // [N*EMB]
    const float* __restrict__ bv,     // [1]
    const float* __restrict__ param,  // [2]
    const float* __restrict__ high,   // [N]
    float* __restrict__ out)          // [B, 2N+1]
{
    __shared__ float red[WAVES][TILE_M];

    const int tid  = threadIdx.x;
    const int lane = tid & 31;
    const int wave = tid >> 5;
    const int b0   = blockIdx.x * TILE_M;

    // ---------------- action head: broadcast rows ----------------
    // a0 = sigmoid(p0)*high[n], a1 = sigmoid(p1)*high[n]*0.5, interleaved
    const float s0 = 1.0f / (1.0f + __expf(-param[0]));
    const float s1 = 0.5f / (1.0f + __expf(-param[1]));
    #pragma unroll 4
    for (int idx = tid; idx < TILE_M * 2 * NGEN; idx += 256) {
        int row = idx >> 10;          // 0..15
        int c   = idx & 1023;         // 0..1023
        float h = high[c >> 1];
        out[(size_t)(b0 + row) * OUT_STRIDE + c] = (c & 1) ? s1 * h : s0 * h;
    }

    // ---------------- value head: fused WMMA f32 16x16x4 ----------------
    const int rowSel = lane & 15;        // batch row within tile (A-matrix M)
    const int kHalf  = (lane >> 4) * 2;  // K sub-offset for upper half-wave
    const int e      = lane & 15;        // emb column for B/C/D fragments

    // Loop-invariant B fragments: W_gen K-slices (16 fragments of K=4)
    v2f bfrag[16];
    #pragma unroll
    for (int k = 0; k < 16; ++k) {
        int K = 4 * k + kHalf;
        bfrag[k][0] = Wg[(K + 0) * EMB + e];
        bfrag[k][1] = Wg[(K + 1) * EMB + e];
    }
    const float bias = bg[e];

    float acc[8] = {0.f, 0.f, 0.f, 0.f, 0.f, 0.f, 0.f, 0.f};

    const float* xrow = x + (size_t)(b0 + rowSel) * NGEN * FGEN + kHalf;

    for (int n = wave; n < NGEN; n += WAVES) {
        const float* ap = xrow + (size_t)n * FGEN;
        const float  wv = Wv[n * EMB + e];

        v8f c = {0.f, 0.f, 0.f, 0.f, 0.f, 0.f, 0.f, 0.f};
        #pragma unroll
        for (int k = 0; k < 16; ++k) {
            v2f a = *(const v2f*)(ap + 4 * k);   // 8B-aligned (kHalf even)
            // D = A(16x4,f32) * B(4x16,f32) + C, chained over K=64
            c = __builtin_amdgcn_wmma_f32_16x16x4_f32(
                    /*neg_a=*/false, a, /*neg_b=*/false, bfrag[k],
                    /*c_mod=*/(short)0, c, /*reuse_a=*/false, /*reuse_b=*/false);
        }
        // prefetch this lane's next n-chunk (streams its row contiguously)
        __builtin_prefetch(ap + WAVES * FGEN, 0, 3);

        // epilogue: bias + relu + scale by W_val, accumulate per (row, e)
        #pragma unroll
        for (int v = 0; v < 8; ++v) {
            float t = c[v] + bias;
            t = t > 0.0f ? t : 0.0f;
            acc[v] = __builtin_fmaf(t, wv, acc[v]);
        }
    }

    // Reduce over the 16 emb lanes within each half-wave (wave32 shfl)
    #pragma unroll
    for (int v = 0; v < 8; ++v) {
        float s = acc[v];
        s += __shfl_xor(s, 1);
        s += __shfl_xor(s, 2);
        s += __shfl_xor(s, 4);
        s += __shfl_xor(s, 8);
        acc[v] = s;
    }
    // lane 0 holds rows 0..7 sums, lane 16 holds rows 8..15 sums
    if ((lane & 15) == 0) {
        int rbase = (lane >> 4) * 8;
        #pragma unroll
        for (int v = 0; v < 8; ++v) red[wave][rbase + v] = acc[v];
    }
    __syncthreads();

    // Cross-wave reduction + final value write: out[b, 1024]
    if (tid < TILE_M) {
        float s = bv[0];
        #pragma unroll
        for (int w = 0; w < WAVES; ++w) s += red[w][tid];
        out[(size_t)(b0 + tid) * OUT_STRIDE + 2 * NGEN] = s;
    }
}

extern "C" void kernel_launch(void* const* d_in, const int* in_sizes, int n_in,
                              void* d_out, int out_size, void* d_ws, size_t ws_size,
                              hipStream_t stream) {
    (void)in_sizes; (void)n_in; (void)d_ws; (void)ws_size; (void)out_size;
    const float* x     = (const float*)d_in[0];  // [4096, 512, 64]
    const float* Wg    = (const float*)d_in[1];  // [64, 16]
    const float* bg    = (const float*)d_in[2];  // [16]
    const float* Wv    = (const float*)d_in[3];  // [8192, 1]
    const float* bv    = (const float*)d_in[4];  // [1]
    const float* param = (const float*)d_in[5];  // [2]
    const float* high  = (const float*)d_in[6];  // [512]
    float* out = (float*)d_out;                  // [4096, 1025]

    dim3 grid(B_TOT / TILE_M);   // 256 blocks, one 16-row batch tile each
    dim3 block(256);             // 8 wave32 waves
    fused_policy_value_kernel<<<grid, block, 0, stream>>>(
        x, Wg, bg, Wv, bv, param, high, out);
}